// HomographySmoothnessLoss_48318382080470
// MI455X (gfx1250) — compile-verified
//
#include <hip/hip_runtime.h>

typedef __attribute__((ext_vector_type(2))) float v2f;
typedef __attribute__((ext_vector_type(8))) float v8f;

#define HH 384
#define WW 640
#define PP (HH * WW)          // 245760 pixels per image
#define BB 2
#define SS 32
#define TILE 256              // pixels per LDS tile
#define VSTRIDE 260           // padded LDS row stride (bank-conflict-free b64)
#define NTILES (PP / TILE)    // 960
#define NB1 120               // k1 blocks per batch (8 tiles each)
#define NB3 240               // k3 blocks per batch
#define EPS2 1e-6f            // EPS^2, EPS=1e-3
#define ALPHA_C 0.45f
#define REG_C 1e-6f
#define MIN_PIX_F 100.0f
#define WEIGHT_C 0.1f

#define RCP_WM1 (1.0f / (float)(WW - 1))
#define RCP_HM1 (1.0f / (float)(HH - 1))

// ---------------------------------------------------------------------------
// Kernel 1: normal-equation accumulation as a WMMA rank-k update.
//   acc[b][seg][col] = sum_p valid[b,seg,p] * M[p,col]
//   M cols: 0:1 1:x 2:y 3:x^2 4:xy 5:y^2 6:fx*x 7:fx*y 8:fx 9:fy*x 10:fy*y 11:fy
// A (16x4 f32) = valid tile (16 segs x 4 pixels), B (4x16 f32) = M rows.
// B-operand values are PRE-COMPUTED per pixel into a column-major LDS tile so
// the WMMA K-loop is pure ds_load_b64 + v_wmma (no divides / no divergence).
// ---------------------------------------------------------------------------
__global__ __launch_bounds__(256) void k1_accum(const float* __restrict__ flow,
                                                const float* __restrict__ mask,
                                                float* __restrict__ accPart,
                                                float* __restrict__ maskPart) {
  __shared__ float vt[SS * VSTRIDE];   // thresholded mask tile (0/1)
  __shared__ float mt[16 * VSTRIDE];   // M-matrix tile, column-major [col][pixel]
  __shared__ float wpart[SS * 8];      // raw-mask per-wave partial sums
  __shared__ float cshare[8 * 256];    // per-wave C tiles for deterministic sum

  const int t = threadIdx.x;
  const int b = blockIdx.y;
  const int bx = blockIdx.x;
  const int wave = t >> 5;
  const int lane = t & 31;
  const int segHalf = wave & 1;        // which 16 segments
  const int quarter = wave >> 1;       // which 64-pixel quarter of the tile

  wpart[t] = 0.0f;                     // 256 == SS*8
  __syncthreads();

  v8f c = {};                          // WMMA accumulator (16 segs x 16 cols)

  const float* maskB = mask + (size_t)b * SS * PP;
  const float* fxG = flow + (size_t)b * 2 * PP;
  const float* fyG = fxG + PP;

  // WMMA operand addressing (constant across tiles)
  const int segBase = segHalf * 16;
  const int colN = lane & 15;          // B: column N ; A: row M (segment)
  const int kHi = (lane >> 4) * 2;     // K sub-pair held by this lane
  const float2* aPtr =
      (const float2*)(vt + (segBase + colN) * VSTRIDE + quarter * 64 + kHi);
  const float2* bPtr =
      (const float2*)(mt + colN * VSTRIDE + quarter * 64 + kHi);

  for (int tile = bx; tile < NTILES; tile += NB1) {
    const int base = tile * TILE;

    // ---- per-pixel M columns (each thread owns one pixel of the tile) ----
    {
      const int gp = base + t;
      const float fx = fxG[gp];
      const float fy = fyG[gp];
      const int row = gp / WW;
      const int col = gp - row * WW;
      const float x = ((float)col * RCP_WM1 - 0.5f) * 2.0f;
      const float y = ((float)row * RCP_HM1 - 0.5f) * 2.0f;
      mt[0 * VSTRIDE + t] = 1.0f;
      mt[1 * VSTRIDE + t] = x;
      mt[2 * VSTRIDE + t] = y;
      mt[3 * VSTRIDE + t] = x * x;
      mt[4 * VSTRIDE + t] = x * y;
      mt[5 * VSTRIDE + t] = y * y;
      mt[6 * VSTRIDE + t] = fx * x;
      mt[7 * VSTRIDE + t] = fx * y;
      mt[8 * VSTRIDE + t] = fx;
      mt[9 * VSTRIDE + t] = fy * x;
      mt[10 * VSTRIDE + t] = fy * y;
      mt[11 * VSTRIDE + t] = fy;
      mt[12 * VSTRIDE + t] = 0.0f;
      mt[13 * VSTRIDE + t] = 0.0f;
      mt[14 * VSTRIDE + t] = 0.0f;
      mt[15 * VSTRIDE + t] = 0.0f;
    }

    // ---- stage mask tile (coalesced per segment row) + raw-mask reduce ----
    for (int i = 0; i < SS; ++i) {
      float m = maskB[(size_t)i * PP + base + t];
      vt[i * VSTRIDE + t] = (m > 0.5f) ? 1.0f : 0.0f;
      float s = m;
      #pragma unroll
      for (int off = 16; off; off >>= 1) s += __shfl_xor(s, off, 32);
      if (lane == 0) wpart[i * 8 + wave] += s;   // lane0-only RMW, no race
    }
    __syncthreads();

    // ---- WMMA chain: 16 K-steps of 4 pixels over this wave's quarter ----
    #pragma unroll 4
    for (int kk = 0; kk < 16; ++kk) {
      const float2 af = aPtr[kk * 2];   // 4 pixels of valid (this lane's K-pair)
      const float2 bf = bPtr[kk * 2];   // matching M-column values
      v2f a;  a.x = af.x;  a.y = af.y;
      v2f bm; bm.x = bf.x; bm.y = bf.y;
      c = __builtin_amdgcn_wmma_f32_16x16x4_f32(false, a, false, bm,
                                                (short)0, c, false, false);
    }
    __syncthreads();   // protect LDS tiles before next iteration's overwrite
  }

  // ---- deterministic cross-wave C reduction via LDS ----
  #pragma unroll
  for (int r = 0; r < 8; ++r) {
    const int m = r + ((lane >> 4) << 3);       // C layout: VGPR r -> M=r / 8+r
    cshare[wave * 256 + m * 16 + (lane & 15)] = c[r];
  }
  __syncthreads();
  for (int idx = t; idx < 512; idx += 256) {
    const int sh = idx >> 8;                    // segHalf
    const int mc = idx & 255;                   // m*16 + col
    float sum = 0.0f;
    #pragma unroll
    for (int q = 0; q < 4; ++q) sum += cshare[(q * 2 + sh) * 256 + mc];
    const int seg = sh * 16 + (mc >> 4);
    const int col = mc & 15;
    accPart[(((size_t)b * NB1 + bx) * SS + seg) * 16 + col] = sum;
  }
  if (t < SS) {
    float sum = 0.0f;
    #pragma unroll
    for (int wv = 0; wv < 8; ++wv) sum += wpart[t * 8 + wv];
    maskPart[((size_t)b * NB1 + bx) * SS + t] = sum;
  }
}

// ---------------------------------------------------------------------------
// Kernel 2: reduce partials, 3x3 symmetric solve -> affine params + flag.
// ---------------------------------------------------------------------------
__global__ __launch_bounds__(64) void k2_solve(const float* __restrict__ accPart,
                                               const float* __restrict__ maskPart,
                                               float* __restrict__ params,
                                               float* __restrict__ masksum) {
  const int t = threadIdx.x;
  if (t < BB * SS) {
    const int b = t >> 5;
    const int s = t & 31;
    float c[12];
    #pragma unroll
    for (int j = 0; j < 12; ++j) c[j] = 0.0f;
    for (int bx = 0; bx < NB1; ++bx) {
      const float* p = accPart + (((size_t)b * NB1 + bx) * SS + s) * 16;
      #pragma unroll
      for (int j = 0; j < 12; ++j) c[j] += p[j];
    }
    // AtA over basis [x,y,1]:
    const float a00 = c[3] + REG_C, a01 = c[4], a02 = c[1];
    const float a11 = c[5] + REG_C, a12 = c[2];
    const float a22 = c[0] + REG_C;
    // adjugate (symmetric) and determinant
    const float i00 = a11 * a22 - a12 * a12;
    const float i01 = a02 * a12 - a01 * a22;
    const float i02 = a01 * a12 - a02 * a11;
    const float i11 = a00 * a22 - a02 * a02;
    const float i12 = a01 * a02 - a00 * a12;
    const float i22 = a00 * a11 - a01 * a01;
    const float det = a00 * i00 + a01 * i01 + a02 * i02;
    const float rdet = 1.0f / det;
    // params = AtA^-1 * Atf   (Atf_x = [c6,c7,c8], Atf_y = [c9,c10,c11])
    float* po = params + (size_t)t * 8;
    po[0] = (i00 * c[6] + i01 * c[7] + i02 * c[8]) * rdet;
    po[1] = (i01 * c[6] + i11 * c[7] + i12 * c[8]) * rdet;
    po[2] = (i02 * c[6] + i12 * c[7] + i22 * c[8]) * rdet;
    po[3] = (i00 * c[9] + i01 * c[10] + i02 * c[11]) * rdet;
    po[4] = (i01 * c[9] + i11 * c[10] + i12 * c[11]) * rdet;
    po[5] = (i02 * c[9] + i12 * c[10] + i22 * c[11]) * rdet;
    po[6] = (c[0] >= MIN_PIX_F) ? 1.0f : 0.0f;  // n_valid gate
    po[7] = 0.0f;
  }
  if (t < SS) {
    float sum = 0.0f;
    for (int i = 0; i < BB * NB1; ++i) sum += maskPart[(size_t)i * SS + t];
    masksum[t] = sum;
  }
}

// ---------------------------------------------------------------------------
// Kernel 3: residual + Charbonnier, soft-mask-weighted per-segment sums.
// ---------------------------------------------------------------------------
__global__ __launch_bounds__(256) void k3_loss(const float* __restrict__ flow,
                                               const float* __restrict__ mask,
                                               const float* __restrict__ params,
                                               float* __restrict__ segPart) {
  __shared__ float sp[SS * 8];
  __shared__ float wp[SS * 8];
  const int t = threadIdx.x;
  const int b = blockIdx.y;
  const int bx = blockIdx.x;
  const int wave = t >> 5;
  const int lane = t & 31;

  sp[t] = params[(size_t)b * SS * 8 + t];   // 256 == SS*8
  __syncthreads();

  float acc[SS];
  #pragma unroll
  for (int s = 0; s < SS; ++s) acc[s] = 0.0f;

  const float* fxG = flow + (size_t)b * 2 * PP;
  const float* fyG = fxG + PP;
  const float* maskB = mask + (size_t)b * SS * PP;

  for (int p = bx * 256 + t; p < PP; p += NB3 * 256) {
    const float fx = fxG[p];
    const float fy = fyG[p];
    const int row = p / WW;
    const int col = p - row * WW;
    const float x = ((float)col * RCP_WM1 - 0.5f) * 2.0f;
    const float y = ((float)row * RCP_HM1 - 0.5f) * 2.0f;
    #pragma unroll
    for (int s = 0; s < SS; ++s) {
      const float* pp = &sp[s * 8];
      const float flag = pp[6];   // residual zeroed when n_valid < MIN_PIX
      const float rx = (fx - (pp[0] * x + pp[1] * y + pp[2])) * flag;
      const float ry = (fy - (pp[3] * x + pp[4] * y + pp[5])) * flag;
      const float lx = __powf(rx * rx + EPS2, ALPHA_C);
      const float ly = __powf(ry * ry + EPS2, ALPHA_C);
      acc[s] += 0.5f * (lx + ly) * maskB[(size_t)s * PP + p];
    }
  }
  #pragma unroll
  for (int s = 0; s < SS; ++s) {
    float v = acc[s];
    #pragma unroll
    for (int off = 16; off; off >>= 1) v += __shfl_xor(v, off, 32);
    if (lane == 0) wp[s * 8 + wave] = v;
  }
  __syncthreads();
  if (t < SS) {
    float sum = 0.0f;
    #pragma unroll
    for (int wv = 0; wv < 8; ++wv) sum += wp[t * 8 + wv];
    segPart[((size_t)b * NB3 + bx) * SS + t] = sum;
  }
}

// ---------------------------------------------------------------------------
// Kernel 4: include-gated final reduction -> scalar loss.
// ---------------------------------------------------------------------------
__global__ __launch_bounds__(32) void k4_final(const float* __restrict__ segPart,
                                               const float* __restrict__ masksum,
                                               float* __restrict__ out) {
  const int t = threadIdx.x;   // 32 threads, one per segment
  float sn = 0.0f;
  for (int i = 0; i < BB * NB3; ++i) sn += segPart[(size_t)i * SS + t];
  const float ms = masksum[t];
  const bool inc = ms >= (MIN_PIX_F * (float)BB);
  // seg_loss * mask_sum == seg_num exactly when included (ms >= 200 >> 1e-6)
  float tl = inc ? sn : 0.0f;
  float tw = inc ? ms : 0.0f;
  #pragma unroll
  for (int off = 16; off; off >>= 1) {
    tl += __shfl_xor(tl, off, 32);
    tw += __shfl_xor(tw, off, 32);
  }
  if (t == 0) out[0] = (tw > 0.0f) ? WEIGHT_C * tl / tw : 0.0f;
}

extern "C" void kernel_launch(void* const* d_in, const int* in_sizes, int n_in,
                              void* d_out, int out_size, void* d_ws, size_t ws_size,
                              hipStream_t stream) {
  const float* flow = (const float*)d_in[0];          // [2,2,384,640]
  const float* segm = (const float*)d_in[1];          // [2,32,384,640]
  float* out = (float*)d_out;                         // scalar

  // workspace layout (floats); every slot fully rewritten each launch
  float* ws = (float*)d_ws;
  float* accPart  = ws;                                           // B*NB1*32*16
  float* maskPart = accPart + (size_t)BB * NB1 * SS * 16;         // B*NB1*32
  float* params   = maskPart + (size_t)BB * NB1 * SS;             // B*32*8
  float* masksum  = params + (size_t)BB * SS * 8;                 // 32
  float* segPart  = masksum + SS;                                 // B*NB3*32

  k1_accum<<<dim3(NB1, BB), 256, 0, stream>>>(flow, segm, accPart, maskPart);
  k2_solve<<<1, 64, 0, stream>>>(accPart, maskPart, params, masksum);
  k3_loss<<<dim3(NB3, BB), 256, 0, stream>>>(flow, segm, params, segPart);
  k4_final<<<1, 32, 0, stream>>>(segPart, masksum, out);
}